// HungarianMatchingLoss_53051436040486
// MI455X (gfx1250) — compile-verified
//
#include <hip/hip_runtime.h>
#include <hip/hip_bf16.h>

// HungarianMatchingLoss for MI455X (gfx1250, wave32).
// K1: per-batch cost matrix via V_WMMA_F32_16X16X4_F32 (one-hot B => bit-exact
//     gather), then Jonker-Volgenant assignment with 64 lanes over columns.
// K2-K4: weighted label-smoothed CE + deterministic tree reductions.

typedef __attribute__((ext_vector_type(2))) float v2f;
typedef __attribute__((ext_vector_type(8))) float v8f;

#define NUM_LAYERS 7
#define BATCH 4096
#define NQ 64
#define NC 10
#define LS 0.1f
#define AUXW 0.5f
#define INF_F 1.0e9f

__global__ __launch_bounds__(64)
void hm_match_kernel(const float* __restrict__ outputs,
                     const float* __restrict__ targets,
                     unsigned char* __restrict__ matched)
{
    const int b    = blockIdx.x;
    const int tid  = threadIdx.x;        // 0..63
    const int lane = tid & 31;
    const int wave = tid >> 5;

    __shared__ float sC[NQ][NQ];         // cost[query][target]
    __shared__ float sprob[NQ][12];      // softmax probs, K padded 10->12 with 0
    __shared__ int   stgt[NQ];           // argmax class per target slot
    __shared__ float su[NQ + 1], sv[NQ + 1], sminv[NQ + 1];
    __shared__ int   sp[NQ + 1], sway[NQ + 1], sused[NQ + 1];
    __shared__ float red_v[NQ];
    __shared__ int   red_i[NQ];

    // ---- per-query: targets argmax (first-max) and final-layer softmax ----
    {
        const float* t = targets + ((size_t)b * NQ + tid) * NC;
        float best = t[0]; int bi = 0;
        #pragma unroll
        for (int c = 1; c < NC; ++c) { float x = t[c]; if (x > best) { best = x; bi = c; } }
        stgt[tid] = bi;

        const float* o = outputs + (((size_t)(NUM_LAYERS - 1) * BATCH + b) * NQ + tid) * NC;
        float xv[NC];
        float m = o[0]; xv[0] = m;
        #pragma unroll
        for (int c = 1; c < NC; ++c) { xv[c] = o[c]; m = fmaxf(m, xv[c]); }
        float s = 0.0f;
        #pragma unroll
        for (int c = 0; c < NC; ++c) { xv[c] = expf(xv[c] - m); s += xv[c]; }
        const float inv = 1.0f / s;
        #pragma unroll
        for (int c = 0; c < NC; ++c) sprob[tid][c] = xv[c] * inv;
        sprob[tid][10] = 0.0f; sprob[tid][11] = 0.0f;
    }
    __syncthreads();

    // ---- cost = (-P)(64x12) x Bt(12x64), B one-hot: 16 tiles, 3 WMMAs each ----
    {
        const int ln16 = lane & 15;
        const int koff = (lane < 16) ? 0 : 2;   // ISA 32-bit A/B frag K split
        for (int tile = wave; tile < 16; tile += 2) {
            const int q0 = (tile >> 2) << 4;
            const int t0 = (tile & 3) << 4;
            const int tcls = stgt[t0 + ln16];
            v8f acc = {0.f, 0.f, 0.f, 0.f, 0.f, 0.f, 0.f, 0.f};
            #pragma unroll
            for (int kk = 0; kk < 12; kk += 4) {
                v2f a, bf;
                a.x  = -sprob[q0 + ln16][kk + koff];
                a.y  = -sprob[q0 + ln16][kk + koff + 1];
                bf.x = (tcls == kk + koff)     ? 1.0f : 0.0f;
                bf.y = (tcls == kk + koff + 1) ? 1.0f : 0.0f;
                acc = __builtin_amdgcn_wmma_f32_16x16x4_f32(
                          false, a, false, bf, (short)0, acc, false, false);
            }
            const int rbase = (lane >> 4) << 3;  // D: VGPR r -> M = r + 8*(lane/16)
            #pragma unroll
            for (int r = 0; r < 8; ++r)
                sC[q0 + rbase + r][t0 + ln16] = acc[r];
        }
    }
    __syncthreads();

    // ---- Jonker-Volgenant (1-indexed, col 0 dummy), lanes own columns ----
    su[tid] = 0.0f; sv[tid] = 0.0f; sp[tid] = 0;
    if (tid == 0) { su[NQ] = 0.0f; sv[NQ] = 0.0f; sp[NQ] = 0; }
    __syncthreads();

    const int j = tid + 1;                       // this thread's column
    for (int i = 1; i <= NQ; ++i) {
        sminv[j] = INF_F; sway[j] = 0; sused[j] = 0;
        if (tid == 0) { sminv[0] = INF_F; sway[0] = 0; sused[0] = 0; sp[0] = i; }
        __syncthreads();

        int j0 = 0;
        for (;;) {
            if (tid == 0) sused[j0] = 1;
            __syncthreads();
            const int  i0    = sp[j0];
            const bool usedj = (sused[j] != 0);
            float minvj = sminv[j];
            if (!usedj) {
                const float cur = sC[i0 - 1][j - 1] - su[i0] - sv[j];
                if (cur < minvj) { minvj = cur; sway[j] = j0; }
            }
            red_v[tid] = usedj ? INF_F : minvj;
            red_i[tid] = j;
            __syncthreads();
            #pragma unroll
            for (int s = 32; s > 0; s >>= 1) {   // argmin, lowest index on tie
                if (tid < s) {
                    const float vo = red_v[tid + s]; const int io = red_i[tid + s];
                    if (vo < red_v[tid] || (vo == red_v[tid] && io < red_i[tid])) {
                        red_v[tid] = vo; red_i[tid] = io;
                    }
                }
                __syncthreads();
            }
            const float delta = red_v[0];
            const int   j1    = red_i[0];
            const int   pj1   = sp[j1];
            if (usedj) { su[sp[j]] += delta; sv[j] -= delta; }
            else       { sminv[j]  = minvj - delta; }
            if (tid == 0) { su[sp[0]] += delta; sv[0] -= delta; }
            __syncthreads();
            j0 = j1;
            if (pj1 == 0) break;
        }
        if (tid == 0) {                          // backtrack augmenting path
            int jj = j0;
            while (jj != 0) { const int jn = sway[jj]; sp[jj] = sp[jn]; jj = jn; }
        }
        __syncthreads();
    }

    // col_ind[p[j]-1] = j-1  =>  matched[p[j]-1] = tgt_ids[j-1]
    matched[(size_t)b * NQ + (sp[j] - 1)] = (unsigned char)stgt[tid];
}

__global__ __launch_bounds__(64)
void hm_loss_kernel(const float* __restrict__ outputs,
                    const unsigned char* __restrict__ matched,
                    const float* __restrict__ cw,
                    float* __restrict__ per_batch)
{
    const int l = blockIdx.y;
    const int b = blockIdx.x;
    const int q = threadIdx.x;

    __shared__ float w[NC];
    __shared__ float r1[NQ], r2[NQ], r3[NQ];
    if (q < NC) w[q] = cw[q];
    __syncthreads();

    const float* x = outputs + ((((size_t)l * BATCH) + b) * NQ + q) * NC;
    float xv[NC];
    float m = x[0]; xv[0] = m;
    #pragma unroll
    for (int c = 1; c < NC; ++c) { xv[c] = x[c]; m = fmaxf(m, xv[c]); }
    float s = 0.0f;
    #pragma unroll
    for (int c = 0; c < NC; ++c) s += expf(xv[c] - m);
    const float lse = m + logf(s);

    float wsum = 0.0f, wdot = 0.0f;
    #pragma unroll
    for (int c = 0; c < NC; ++c) { wsum += w[c]; wdot += w[c] * xv[c]; }

    const int   y      = (int)matched[(size_t)b * NQ + q];
    const float nll    = lse - xv[y];
    const float wy     = w[y];
    const float smooth = lse * wsum - wdot;    // -(logp*w).sum

    r1[q] = wy * nll; r2[q] = smooth; r3[q] = wy;
    __syncthreads();
    #pragma unroll
    for (int st = 32; st > 0; st >>= 1) {
        if (q < st) { r1[q] += r1[q + st]; r2[q] += r2[q + st]; r3[q] += r3[q + st]; }
        __syncthreads();
    }
    if (q == 0)
        per_batch[(size_t)l * BATCH + b] =
            ((1.0f - LS) * r1[0] + (LS / (float)NC) * r2[0]) / r3[0];
}

__global__ __launch_bounds__(256)
void hm_layer_reduce(const float* __restrict__ per_batch,
                     float* __restrict__ layer_mean)
{
    const int l = blockIdx.x;
    const int t = threadIdx.x;
    __shared__ float s[256];
    float acc = 0.0f;
    for (int b = t; b < BATCH; b += 256)        // fixed order: deterministic
        acc += per_batch[(size_t)l * BATCH + b];
    s[t] = acc;
    __syncthreads();
    for (int st = 128; st > 0; st >>= 1) {
        if (t < st) s[t] += s[t + st];
        __syncthreads();
    }
    if (t == 0) layer_mean[l] = s[0] / (float)BATCH;
}

__global__ void hm_finalize(const float* __restrict__ lm, float* __restrict__ out)
{
    float aux = 0.0f;
    for (int l = 0; l < NUM_LAYERS - 1; ++l) aux += lm[l];
    out[0] = lm[NUM_LAYERS - 1] + AUXW * (aux / (float)(NUM_LAYERS - 1));
}

extern "C" void kernel_launch(void* const* d_in, const int* in_sizes, int n_in,
                              void* d_out, int out_size, void* d_ws, size_t ws_size,
                              hipStream_t stream) {
    (void)in_sizes; (void)n_in; (void)out_size; (void)ws_size;
    const float* outputs = (const float*)d_in[0];   // (7,4096,64,10) f32
    const float* targets = (const float*)d_in[1];   // (4096,64,10)  f32
    const float* cw      = (const float*)d_in[2];   // (10,)         f32
    float* out = (float*)d_out;

    char* ws = (char*)d_ws;
    unsigned char* matched    = (unsigned char*)ws;                    // 256 KB
    float*         per_batch  = (float*)(ws + 512 * 1024);             // 112 KB
    float*         layer_mean = (float*)(ws + 512 * 1024 + 128 * 1024);

    hm_match_kernel<<<BATCH, 64, 0, stream>>>(outputs, targets, matched);
    dim3 g(BATCH, NUM_LAYERS);
    hm_loss_kernel<<<g, 64, 0, stream>>>(outputs, matched, cw, per_batch);
    hm_layer_reduce<<<NUM_LAYERS, 256, 0, stream>>>(per_batch, layer_mean);
    hm_finalize<<<1, 1, 0, stream>>>(layer_mean, out);
}